// ChainLoss_72567767433601
// MI455X (gfx1250) — compile-verified
//
#include <hip/hip_runtime.h>
#include <cstdint>
#include <cstddef>

// ---------------------------------------------------------------------------
// ChainLoss (LF-MMI denominator forward recursion + numerator gather)
//   B=64 blocks (one per batch element), 1024 threads (32 waves) each.
//   LDS: obs row double-buffer (2x12KB, async-prefetched via
//   global_load_async_to_lds_b128 / s_wait_asynccnt), alpha/alpha_new/init
//   (3x8KB), reduction scratch.  Arc stream (640KB) stays L2-resident.
//   Wave32 butterfly reductions via ds_swizzle; scatter via ds_add_f32.
// ---------------------------------------------------------------------------

#define BB   64
#define TT   300
#define PP   3072
#define SS   2000
#define AA   40000
#define LEAKY 0.1f
#define NTHR 1024
#define NWAVES (NTHR / 32)

// ----- CDNA5 async global->LDS helpers -------------------------------------
__device__ __forceinline__ void async_row_load(const float* __restrict__ gsrc,
                                               float* lds_dst, int tid) {
  // One 16B chunk per lane; 768 chunks cover one 3072-float row.
  if (tid < PP / 4) {
#if defined(__HIP_DEVICE_COMPILE__) && defined(__gfx1250__)
    uint32_t lds_addr = (uint32_t)(uintptr_t)(lds_dst + tid * 4); // low 32 bits = LDS offset
    uint32_t voff     = (uint32_t)tid * 16u;                      // byte offset vgpr
    uint64_t gbase    = (uint64_t)(uintptr_t)gsrc;                // sgpr pair base
    asm volatile("global_load_async_to_lds_b128 %0, %1, %2"
                 :: "v"(lds_addr), "v"(voff), "s"(gbase) : "memory");
#else
    ((float4*)lds_dst)[tid] = ((const float4*)gsrc)[tid];
#endif
  }
}

__device__ __forceinline__ void async_wait() {
#if defined(__HIP_DEVICE_COMPILE__) && defined(__gfx1250__)
  asm volatile("s_wait_asynccnt 0x0" ::: "memory");
#endif
}

// ----- wave32 butterfly all-reduce via ds_swizzle (lane ^ mask) ------------
__device__ __forceinline__ float wave_allreduce(float v) {
  // group-of-32 swizzle: imm = (xor_mask << 10) | and_mask(0x1F)
  v += __int_as_float(__builtin_amdgcn_ds_swizzle(__float_as_int(v), 0x041F)); // ^1
  v += __int_as_float(__builtin_amdgcn_ds_swizzle(__float_as_int(v), 0x081F)); // ^2
  v += __int_as_float(__builtin_amdgcn_ds_swizzle(__float_as_int(v), 0x101F)); // ^4
  v += __int_as_float(__builtin_amdgcn_ds_swizzle(__float_as_int(v), 0x201F)); // ^8
  v += __int_as_float(__builtin_amdgcn_ds_swizzle(__float_as_int(v), 0x401F)); // ^16
  return v;
}

__device__ __forceinline__ float block_reduce_sum(float v, float* red, int tid) {
  v = wave_allreduce(v);
  const int wave = tid >> 5, lane = tid & 31;
  if (lane == 0) red[wave] = v;
  __syncthreads();
  if (wave == 0) {
    float w = wave_allreduce(red[lane]);   // NWAVES == 32, all slots valid
    if (lane == 0) red[0] = w;
  }
  __syncthreads();
  float r = red[0];
  __syncthreads();                          // allow scratch reuse
  return r;
}

// ----- main kernel: one block per batch element ----------------------------
__global__ __launch_bounds__(NTHR) void chain_fwd_kernel(
    const float* __restrict__ x,          // [B,T,P]
    const int*   __restrict__ targets,    // [B,T]
    const int*   __restrict__ arc_from,   // [A]
    const int*   __restrict__ arc_to,     // [A]
    const int*   __restrict__ arc_pdf,    // [A]
    const float* __restrict__ arc_w,      // [A]
    const float* __restrict__ init_probs, // [S]
    const float* __restrict__ final_probs,// [S]
    float*       __restrict__ block_out)  // [B][2] : {num_partial, logz}
{
  __shared__ __align__(16) float s_obs[2][PP];   // raw x prefetch -> exp in place
  __shared__ float s_alpha[SS];
  __shared__ float s_alphan[SS];
  __shared__ float s_init[SS];
  __shared__ float s_red[NWAVES];

  const int tid = threadIdx.x;
  const int b   = blockIdx.x;
  const float* xb = x + (size_t)b * TT * PP;

  // --- normalized initial distribution into LDS; alpha0 = init -------------
  float ip = 0.f;
  for (int s = tid; s < SS; s += NTHR) { float v = init_probs[s]; s_init[s] = v; ip += v; }
  const float isum = block_reduce_sum(ip, s_red, tid);
  const float inv_isum = 1.0f / isum;
  for (int s = tid; s < SS; s += NTHR) {
    float v = s_init[s] * inv_isum;
    s_init[s]   = v;
    s_alpha[s]  = v;
    s_alphan[s] = 0.f;
  }
  // prefetch obs row 0 (async -> LDS, bypasses VGPRs)
  async_row_load(xb, &s_obs[0][0], tid);
  __syncthreads();

  const int4*   fv = (const int4*)arc_from;
  const int4*   tv = (const int4*)arc_to;
  const int4*   pv = (const int4*)arc_pdf;
  const float4* wv = (const float4*)arc_w;

  float logz   = 0.f;   // same value on all threads
  float numacc = 0.f;   // meaningful on tid 0 only

  for (int t = 0; t < TT; ++t) {
    float* obs = s_obs[t & 1];

    // complete this wave's async loads, then make buffer visible to all waves
    async_wait();
    __syncthreads();

    // exp(clip(x)) in place, vectorized: one b128 LDS ld/st per thread
    if (tid < PP / 4) {
      float4 v = ((float4*)obs)[tid];
      v.x = __expf(fminf(fmaxf(v.x, -30.f), 30.f));
      v.y = __expf(fminf(fmaxf(v.y, -30.f), 30.f));
      v.z = __expf(fminf(fmaxf(v.z, -30.f), 30.f));
      v.w = __expf(fminf(fmaxf(v.w, -30.f), 30.f));
      ((float4*)obs)[tid] = v;
    }
    __syncthreads();

    // numerator contribution: exp(x)[b,t,target]
    if (tid == 0) numacc += obs[targets[b * TT + t]];

    // prefetch next row while arcs are processed
    if (t + 1 < TT) async_row_load(xb + (size_t)(t + 1) * PP, &s_obs[(t + 1) & 1][0], tid);

    // --- arc scatter: alpha_new[to] += alpha[from] * obs[pdf] * w ----------
    for (int p = tid; p < AA / 4; p += NTHR) {
      const int4   f = fv[p];
      const int4   d = tv[p];
      const int4   q = pv[p];
      const float4 w = wv[p];
      atomicAdd(&s_alphan[d.x], s_alpha[f.x] * obs[q.x] * w.x);
      atomicAdd(&s_alphan[d.y], s_alpha[f.y] * obs[q.y] * w.y);
      atomicAdd(&s_alphan[d.z], s_alpha[f.z] * obs[q.z] * w.z);
      atomicAdd(&s_alphan[d.w], s_alpha[f.w] * obs[q.w] * w.w);
    }
    __syncthreads();

    // --- leaky-HMM mix + renormalize (fused; Σinit==1 => scale=tot*(1+L)) --
    float pt = 0.f;
    for (int s = tid; s < SS; s += NTHR) pt += s_alphan[s];
    const float tot   = block_reduce_sum(pt, s_red, tid);
    const float leak  = LEAKY * tot;
    const float scale = tot * (1.0f + LEAKY);
    const float inv   = 1.0f / scale;
    for (int s = tid; s < SS; s += NTHR) {
      s_alpha[s]  = (s_alphan[s] + leak * s_init[s]) * inv;
      s_alphan[s] = 0.f;
    }
    logz += __logf(scale);
    __syncthreads();
  }

  // --- final-prob termination ----------------------------------------------
  float pf = 0.f;
  for (int s = tid; s < SS; s += NTHR) pf += s_alpha[s] * final_probs[s];
  const float fin = block_reduce_sum(pf, s_red, tid);
  logz += __logf(fin);

  if (tid == 0) {
    block_out[b * 2 + 0] = numacc;
    block_out[b * 2 + 1] = logz;
  }
}

// ----- deterministic final reduction ---------------------------------------
__global__ void chain_finalize_kernel(const float* __restrict__ block_out,
                                      float* __restrict__ out) {
  if (threadIdx.x == 0 && blockIdx.x == 0) {
    float num = 0.f, den = 0.f;
    for (int b = 0; b < BB; ++b) {      // fixed order -> deterministic
      num += block_out[b * 2 + 0];
      den += block_out[b * 2 + 1];
    }
    out[0] = -(num - den) / (float)(BB * TT);
  }
}

extern "C" void kernel_launch(void* const* d_in, const int* in_sizes, int n_in,
                              void* d_out, int out_size, void* d_ws, size_t ws_size,
                              hipStream_t stream) {
  (void)in_sizes; (void)n_in; (void)out_size; (void)ws_size;
  const float* x           = (const float*)d_in[0];
  const int*   targets     = (const int*)  d_in[1];
  const int*   arc_from    = (const int*)  d_in[2];
  const int*   arc_to      = (const int*)  d_in[3];
  const int*   arc_pdf     = (const int*)  d_in[4];
  const float* arc_w       = (const float*)d_in[5];
  const float* init_probs  = (const float*)d_in[6];
  const float* final_probs = (const float*)d_in[7];
  float* block_out = (float*)d_ws;     // [B][2] floats, fully written each call

  chain_fwd_kernel<<<dim3(BB), dim3(NTHR), 0, stream>>>(
      x, targets, arc_from, arc_to, arc_pdf, arc_w, init_probs, final_probs,
      block_out);
  chain_finalize_kernel<<<dim3(1), dim3(64), 0, stream>>>(block_out, (float*)d_out);
}